// DWTModelFullBand_39943195853364
// MI455X (gfx1250) — compile-verified
//
#include <hip/hip_runtime.h>
#include <cstdint>

// DWT(2-level) -> IDWT(2-level) Haar roundtrip on (32,3,512,512) f32.
// Block-local on 4x4 tiles -> one thread per tile. Memory-bound:
// ~201 MB traffic @ 23.3 TB/s => ~8.6 us floor, ~1.25 flop/byte => no WMMA.
// CDNA5 paths used:
//   - GLOBAL_LOAD_ASYNC_TO_LDS_B128 (ASYNCcnt-tracked DMA) for the inbound
//     stream; single s_wait_asynccnt 0 covers the strip.
//   - Nontemporal B128 stores so the 100 MB input stays resident in the
//     192 MB L2 across graph replays (input+output = 201 MB > L2, so the
//     output stream must not be cached).
// Grid exactly covers the tile count (1,572,864 = 6144 * 256), so the kernel
// is branch-free: no saveexec diamond, loads -> wait -> readback is a single
// straight-line sequence.

#define THREADS 256
#define PLANE_W 512          // row stride in floats
typedef float v4f __attribute__((ext_vector_type(4)));

__device__ __forceinline__ void haar_fwd(float a, float b, float c, float d,
                                         float& ll, float& lh, float& hl, float& hh) {
    // match reference's left-to-right association exactly
    ll = (((a + b) + c) + d) * 0.5f;
    lh = (((c + d) - a) - b) * 0.5f;
    hl = (((b + d) - a) - c) * 0.5f;
    hh = (((a + d) - b) - c) * 0.5f;
}

__device__ __forceinline__ void haar_inv(float ll, float lh, float hl, float hh,
                                         float& a, float& b, float& c, float& d) {
    a = (((ll - lh) - hl) + hh) * 0.5f;
    b = (((ll - lh) + hl) - hh) * 0.5f;
    c = (((ll + lh) - hl) - hh) * 0.5f;
    d = (((ll + lh) + hl) + hh) * 0.5f;
}

__global__ __launch_bounds__(THREADS)
void dwt_roundtrip_kernel(const float* __restrict__ x, float* __restrict__ y) {
    // Staging buffer filled by the async DMA path. Layout buf[row][tid] keeps
    // both the async writes and ds_load_b128 readback bank-conflict-free
    // (wave32 x b128 = 128 words = the 2-pass minimum over 64 banks).
    __shared__ v4f buf[4][THREADS];

    const int t = blockIdx.x * THREADS + threadIdx.x;

    // 128x128 tiles per 512x512 plane; consecutive t => consecutive tileX
    // => consecutive 16B chunks => fully coalesced B128 traffic.
    const int plane = t >> 14;
    const int tp    = t & 16383;
    const int ty    = tp >> 7;
    const int tx    = tp & 127;

    const size_t base = ((size_t)plane << 18) + (size_t)(ty * 4) * PLANE_W + (size_t)(tx * 4);
    const float* src = x + base;

#pragma unroll
    for (int r = 0; r < 4; ++r) {
        // AS3 offsets are the low 32 bits of the generic LDS address.
        unsigned lds_off = (unsigned)(uintptr_t)&buf[r][threadIdx.x];
        unsigned long long ga = (unsigned long long)(uintptr_t)(src + (size_t)r * PLANE_W);
        asm volatile("global_load_async_to_lds_b128 %0, %1, off"
                     :: "v"(lds_off), "v"(ga)
                     : "memory");
    }

#if __has_builtin(__builtin_amdgcn_s_wait_asynccnt)
    __builtin_amdgcn_s_wait_asynccnt(0);
#else
    asm volatile("s_wait_asynccnt 0" ::: "memory");
#endif

    // Pull the 4x4 tile out of LDS (each thread reads only its own slots; the
    // wave-level asynccnt wait above is the only sync needed).
    float v[4][4];
#pragma unroll
    for (int r = 0; r < 4; ++r) {
        v4f q = buf[r][threadIdx.x];
        v[r][0] = q.x; v[r][1] = q.y; v[r][2] = q.z; v[r][3] = q.w;
    }

    // Level-1 forward DWT on the four 2x2 blocks.
    float ll1[2][2], lh1[2][2], hl1[2][2], hh1[2][2];
#pragma unroll
    for (int i = 0; i < 2; ++i)
#pragma unroll
        for (int j = 0; j < 2; ++j)
            haar_fwd(v[2*i][2*j], v[2*i][2*j+1], v[2*i+1][2*j], v[2*i+1][2*j+1],
                     ll1[i][j], lh1[i][j], hl1[i][j], hh1[i][j]);

    // Level-2 forward DWT on LL1 (single 2x2 block).
    float ll2, lh2, hl2, hh2;
    haar_fwd(ll1[0][0], ll1[0][1], ll1[1][0], ll1[1][1], ll2, lh2, hl2, hh2);

    // Level-2 inverse.
    float llr[2][2];
    haar_inv(ll2, lh2, hl2, hh2, llr[0][0], llr[0][1], llr[1][0], llr[1][1]);

    // Level-1 inverse -> reconstructed 4x4 tile.
    float o[4][4];
#pragma unroll
    for (int i = 0; i < 2; ++i)
#pragma unroll
        for (int j = 0; j < 2; ++j)
            haar_inv(llr[i][j], lh1[i][j], hl1[i][j], hh1[i][j],
                     o[2*i][2*j], o[2*i][2*j+1], o[2*i+1][2*j], o[2*i+1][2*j+1]);

    // Nontemporal B128 stores: keep the output stream out of L2 so the input
    // (100 MB < 192 MB L2) stays resident for graph-replay re-reads.
    float* dst = y + base;
#pragma unroll
    for (int r = 0; r < 4; ++r) {
        v4f q = { o[r][0], o[r][1], o[r][2], o[r][3] };
        __builtin_nontemporal_store(q, (v4f*)(dst + (size_t)r * PLANE_W));
    }
}

extern "C" void kernel_launch(void* const* d_in, const int* in_sizes, int n_in,
                              void* d_out, int out_size, void* d_ws, size_t ws_size,
                              hipStream_t stream) {
    const float* x = (const float*)d_in[0];
    float* y = (float*)d_out;

    const int n = in_sizes[0];           // 32*3*512*512 = 25,165,824 floats
    const int n_tiles = n / 16;          // one thread per 4x4 tile
    const int blocks = n_tiles / THREADS; // exact: 6144 blocks for this shape

    dwt_roundtrip_kernel<<<blocks, THREADS, 0, stream>>>(x, y);
}